// EncoderSmoothedMaxPoolingLoss_9285719294294
// MI455X (gfx1250) — compile-verified
//
#include <hip/hip_runtime.h>
#include <math.h>

// EncoderSmoothedMaxPoolingLoss for MI455X (gfx1250, wave32).
// Inputs (setup_inputs order):
//   d_in[0] X            float32 [B, F, C]
//   d_in[1] lengths      int32   [B]
//   d_in[2] tgt          int32   [B, P]
//   d_in[3] p_end        int32   [B]
//   d_in[4] phoneme_nums int32   [B]
// Output: d_out[0] = scalar float32 loss.

#define WIN     20
#define OFFS    40
#define TAPS    9
#define SIGMA   4.0f

typedef __attribute__((ext_vector_type(2))) float v2f;
typedef __attribute__((ext_vector_type(8))) float v8f;

// ---------------------------------------------------------------------------
// Wave-level sum using V_WMMA_F32_16X16X4_F32.
// With B = all-ones and C = 0:  sum over all 256 D entries == 16 * sum(A).
// Each lane contributes {s, 0} as its A fragment, so sum(A) == wave sum of s.
// This identity is independent of the exact lane<->(m,k) layout, hence robust.
// Exact /16 (power of two) recovers the wave sum. Deterministic.
// ---------------------------------------------------------------------------
__device__ __forceinline__ float wave_reduce_sum(float v) {
#if defined(__gfx1250__)
    v2f a;    a[0] = v;    a[1] = 0.0f;
    v2f ones; ones[0] = 1.0f; ones[1] = 1.0f;
    v8f c = {};
    v8f d = __builtin_amdgcn_wmma_f32_16x16x4_f32(
        /*neg_a=*/false, a, /*neg_b=*/false, ones,
        /*c_mod=*/(short)0, c, /*reuse_a=*/false, /*reuse_b=*/false);
    float t = d[0] + d[1] + d[2] + d[3] + d[4] + d[5] + d[6] + d[7];
    #pragma unroll
    for (int off = 16; off >= 1; off >>= 1)
        t += __shfl_xor(t, off, 32);
    return t * 0.0625f;   // / 16, exact
#else
    #pragma unroll
    for (int off = 16; off >= 1; off >>= 1)
        v += __shfl_xor(v, off, 32);
    return v;
#endif
}

// Deterministic block reduction (<= 32 waves). Result valid in thread 0.
__device__ __forceinline__ float block_reduce_sum(float v, float* lds) {
    const int lane = threadIdx.x & 31;
    const int wid  = threadIdx.x >> 5;
    float w = wave_reduce_sum(v);      // uniform call: EXEC all ones for WMMA
    if (lane == 0) lds[wid] = w;
    __syncthreads();
    float total = 0.0f;
    if (threadIdx.x == 0) {
        const int nwaves = (blockDim.x + 31) >> 5;
        for (int i = 0; i < nwaves; ++i) total += lds[i];
    }
    return total;
}

__device__ __forceinline__ void make_filter(float* g) {
    float s = 0.0f;
    #pragma unroll
    for (int j = 0; j < TAPS; ++j) {
        const float x = (float)(j - TAPS / 2) / SIGMA;
        g[j] = expf(-0.5f * x * x);
        s += g[j];
    }
    #pragma unroll
    for (int j = 0; j < TAPS; ++j) g[j] /= s;
}

// ---------------------------------------------------------------------------
// Kernel 1: per-phoneme smoothed-max loss.  grid = B blocks, block = P threads.
// Thread p of block b handles phoneme (b, p): gather 20 frames of channel
// tgt[b,p], 9-tap smooth (window t-3 .. t+5, matching the reference's
// lpad=9 / center=10 cross-correlation), masked max, contribute -max.
// ---------------------------------------------------------------------------
__global__ void phoneme_loss_kernel(const float* __restrict__ X,
                                    const int* __restrict__ lengths,
                                    const int* __restrict__ tgt,
                                    const int* __restrict__ p_end,
                                    const int* __restrict__ phoneme_nums,
                                    float* __restrict__ partials,
                                    int F, int C, int P) {
    const int b = blockIdx.x;
    const int p = threadIdx.x;

    const int length  = lengths[b];
    const int pn      = phoneme_nums[b];
    const int ends_ph = min(p_end[b] + OFFS, length);

    float g[TAPS];
    make_filter(g);

    float contrib = 0.0f;
    if (p < pn) {
        int start = ends_ph - WIN * (pn - p);
        start = max(start, 0);
        const int end  = min(start + WIN, length);
        const int wlen = end - start;                 // valid frames in window
        const int ch   = tgt[b * P + p];

        const float* Xbc = X + (size_t)b * F * C + ch;
        float vals[WIN];
        #pragma unroll
        for (int t = 0; t < WIN; ++t) {
            // t < wlen implies start+t < end <= length <= F: in-bounds read.
            vals[t] = (t < wlen) ? Xbc[(size_t)(start + t) * C] : 0.0f;
        }

        float m = -INFINITY;
        #pragma unroll
        for (int t = 0; t < WIN; ++t) {
            if (t < wlen) {
                float acc = 0.0f;
                #pragma unroll
                for (int j = 0; j < TAPS; ++j) {
                    const int idx = t + j - 3;        // reference's +1 shift
                    const float v = (idx >= 0 && idx < WIN) ? vals[idx] : 0.0f;
                    acc = fmaf(g[j], v, acc);
                }
                m = fmaxf(m, acc);
            }
        }
        contrib = -m;
    }

    __shared__ float lds[32];
    const float total = block_reduce_sum(contrib, lds);
    if (threadIdx.x == 0) partials[b] = total;
}

// ---------------------------------------------------------------------------
// Kernel 2: silence-frame sum of X[b, f, 0].
// grid = B * CHUNKS blocks; each block covers a contiguous frame chunk.
// ---------------------------------------------------------------------------
__global__ void silence_loss_kernel(const float* __restrict__ X,
                                    const int* __restrict__ lengths,
                                    const int* __restrict__ p_end,
                                    const int* __restrict__ phoneme_nums,
                                    float* __restrict__ partials,
                                    int F, int C, int chunks) {
    const int b     = blockIdx.x / chunks;
    const int chunk = blockIdx.x % chunks;

    const int length  = lengths[b];
    const int pn      = phoneme_nums[b];
    const int ends_ph = min(p_end[b] + OFFS, length);

    const int first_start = max(ends_ph - WIN * pn, 0);             // starts[b,0]
    const int last_end    = min(max(ends_ph - WIN, 0) + WIN, length); // ends[b,pn-1]

    const int fpc = (F + chunks - 1) / chunks;
    const int f0  = chunk * fpc;
    const int f1  = min(f0 + fpc, F);

    float s = 0.0f;
    for (int f = f0 + (int)threadIdx.x; f < f1; f += (int)blockDim.x) {
        const bool sil = (f < first_start) || ((f >= last_end) && (f < length));
        if (sil) s += X[((size_t)b * F + f) * C];   // channel SIL == 0
    }

    __shared__ float lds[32];
    const float total = block_reduce_sum(s, lds);
    if (threadIdx.x == 0) partials[b * chunks + chunk] = total;
}

// ---------------------------------------------------------------------------
// Kernel 3: fixed-order finalize (deterministic). loss = sum(ph) - sum(sil).
// ---------------------------------------------------------------------------
__global__ void finalize_kernel(const float* __restrict__ ph_partials, int n_ph,
                                const float* __restrict__ sil_partials, int n_sil,
                                float* __restrict__ out) {
    if (blockIdx.x == 0 && threadIdx.x == 0) {
        double acc = 0.0;
        for (int i = 0; i < n_ph; ++i)  acc += (double)ph_partials[i];
        for (int i = 0; i < n_sil; ++i) acc -= (double)sil_partials[i];
        out[0] = (float)acc;
    }
}

extern "C" void kernel_launch(void* const* d_in, const int* in_sizes, int n_in,
                              void* d_out, int out_size, void* d_ws, size_t ws_size,
                              hipStream_t stream) {
    const float* X    = (const float*)d_in[0];
    const int* lengths       = (const int*)d_in[1];
    const int* tgt           = (const int*)d_in[2];
    const int* p_end         = (const int*)d_in[3];
    const int* phoneme_nums  = (const int*)d_in[4];
    float* out = (float*)d_out;

    const int B = in_sizes[1];
    const int P = in_sizes[2] / B;
    const int C = 256;                        // reference channel count
    const int F = (int)((long long)in_sizes[0] / ((long long)B * C));

    const int CHUNKS = 8;                     // silence-sum parallelism per batch

    float* ph_partials  = (float*)d_ws;       // [B]
    float* sil_partials = ph_partials + B;    // [B * CHUNKS]

    phoneme_loss_kernel<<<B, P, 0, stream>>>(
        X, lengths, tgt, p_end, phoneme_nums, ph_partials, F, C, P);

    silence_loss_kernel<<<B * CHUNKS, 256, 0, stream>>>(
        X, lengths, p_end, phoneme_nums, sil_partials, F, C, CHUNKS);

    finalize_kernel<<<1, 32, 0, stream>>>(
        ph_partials, B, sil_partials, B * CHUNKS, out);
}